// GCN_21887153340598
// MI455X (gfx1250) — compile-verified
//
#include <hip/hip_runtime.h>
#include <hip/hip_bf16.h>

typedef float v2f __attribute__((ext_vector_type(2)));
typedef float v4f __attribute__((ext_vector_type(4)));
typedef float v8f __attribute__((ext_vector_type(8)));

// ---------------------------------------------------------------------------
// Kernel 1: build CSR row_ptr from sorted edge_row via per-node binary search.
// row_ptr[i] = lower_bound(edge_row, i), row_ptr[n_nodes] = n_edges.
// ---------------------------------------------------------------------------
__global__ __launch_bounds__(256) void gcn_rowptr_kernel(
    const int* __restrict__ edge_row, int* __restrict__ row_ptr,
    int n_nodes, int n_edges) {
  int i = blockIdx.x * blockDim.x + threadIdx.x;
  if (i > n_nodes) return;
  if (i == n_nodes) { row_ptr[i] = n_edges; return; }
  int lo = 0, hi = n_edges;
  while (lo < hi) {
    int mid = (lo + hi) >> 1;
    if (edge_row[mid] < i) lo = mid + 1; else hi = mid;
  }
  row_ptr[i] = lo;
}

// ---------------------------------------------------------------------------
// Kernel 2/4: dense GEMM  C[M x N] = A[M x 128] * B[128 x N]
// using V_WMMA_F32_16X16X4_F32 (wave32, 16x16 f32 tile, K-step 4).
//
// Per-lane fragment layout (CDNA5 ISA 7.12.2, 32-bit A 16x4):
//   lane = 16*half + l16 ;  A frag VGPR0/1 = A[m, k+2*half +{0,1}]  -> float2
//   B frag (4x16, rows striped across lanes): B[k+2*half +{0,1}, n] -> float2
//   C/D:   VGPR v, row = v + 8*half, col = l16
//
// B is staged in LDS pre-interleaved as K-pairs so the B fragment is a single
// ds_load_b64 per WMMA. One wave owns one 16-wide N tile; M tiles are skipped
// wave-uniformly (EXEC stays all-ones as WMMA requires; M % 16 == 0 here).
// ---------------------------------------------------------------------------
template <int N>
__global__ __launch_bounds__(256) void gcn_gemm_wmma_kernel(
    const float* __restrict__ A, const float* __restrict__ B,
    float* __restrict__ C, int M) {
  constexpr int K = 128;
  constexpr int NT = N / 16;       // N tiles per block row-group
  constexpr int RG = 8 / NT;       // row-groups among the 8 waves
  constexpr int M_BLOCK = 64;
  constexpr int M_ITERS = M_BLOCK / (RG * 16);

  // LDS holds B as float2 pairs: ldsB[p*N + n] = {B[2p][n], B[2p+1][n]}
  __shared__ v2f ldsB[(K / 2) * N];

  const int tid = threadIdx.x;
  for (int i = tid; i < (K / 2) * N; i += 256) {
    const int p = i / N;
    const int n = i % N;
    v2f t;
    t.x = B[(2 * p) * N + n];
    t.y = B[(2 * p + 1) * N + n];
    ldsB[i] = t;
  }
  __syncthreads();

  const int wave = tid >> 5;
  const int lane = tid & 31;
  const int half = lane >> 4;   // 0: lanes 0-15, 1: lanes 16-31
  const int l16  = lane & 15;
  const int nbase = (wave % NT) * 16;
  const int rg    = wave / NT;
  const int mblock = blockIdx.x * M_BLOCK;

  for (int mi = 0; mi < M_ITERS; ++mi) {
    const int mbase = mblock + (rg + mi * RG) * 16;
    if (mbase >= M) continue;  // wave-uniform skip, EXEC stays full

    v8f acc = {};
    const float* Arow = A + (size_t)(mbase + l16) * K + 2 * half;
    const v2f* Bcol = ldsB + half * N + nbase + l16;

#pragma unroll
    for (int k = 0; k < K; k += 4) {
      v2f a = *(const v2f*)(Arow + k);             // A[m][k+2h .. k+2h+1]
      v2f b = Bcol[(k >> 1) * N];                  // {B[k+2h][n], B[k+2h+1][n]}
      acc = __builtin_amdgcn_wmma_f32_16x16x4_f32(
          /*neg_a=*/false, a, /*neg_b=*/false, b,
          /*c_mod=*/(short)0, acc, /*reuse_a=*/false, /*reuse_b=*/false);
    }

#pragma unroll
    for (int v = 0; v < 8; ++v) {
      C[(size_t)(mbase + v + 8 * half) * N + nbase + l16] = acc[v];
    }
  }
}

// ---------------------------------------------------------------------------
// Kernel 3/5: SpMM  out[i,:] = sum_{e in [rp[i],rp[i+1])} w[e] * feat[col[e],:]
// One wave per node; each lane owns F/32 contiguous floats of the row so the
// gather is a single global_load_b128 (F=128) / b64 (F=64) per lane per edge.
// Two-edge unroll with independent accumulators for ILP. No atomics needed.
// ---------------------------------------------------------------------------
template <int F, bool RELU>
__global__ __launch_bounds__(256) void gcn_spmm_kernel(
    const float* __restrict__ feat, const int* __restrict__ row_ptr,
    const int* __restrict__ edge_col, const float* __restrict__ edge_w,
    float* __restrict__ out, int n_nodes) {
  constexpr int VPL = F / 32;  // floats per lane (4 or 2)
  const int wave = threadIdx.x >> 5;
  const int lane = threadIdx.x & 31;
  const int node = blockIdx.x * 8 + wave;
  if (node >= n_nodes) return;

  const int e0 = row_ptr[node];
  const int e1 = row_ptr[node + 1];

  float acc0[VPL], acc1[VPL];
#pragma unroll
  for (int i = 0; i < VPL; ++i) { acc0[i] = 0.f; acc1[i] = 0.f; }

  int e = e0;
  for (; e + 1 < e1; e += 2) {
    const int   c0 = edge_col[e];
    const float w0 = edge_w[e];
    const int   c1 = edge_col[e + 1];
    const float w1 = edge_w[e + 1];
    const float* s0 = feat + (size_t)c0 * F + lane * VPL;
    const float* s1 = feat + (size_t)c1 * F + lane * VPL;
    if (VPL == 4) {
      v4f a = *(const v4f*)s0;
      v4f b = *(const v4f*)s1;
      acc0[0] += w0 * a.x; acc0[1] += w0 * a.y;
      acc0[2] += w0 * a.z; acc0[3] += w0 * a.w;
      acc1[0] += w1 * b.x; acc1[1] += w1 * b.y;
      acc1[2] += w1 * b.z; acc1[3] += w1 * b.w;
    } else {
      v2f a = *(const v2f*)s0;
      v2f b = *(const v2f*)s1;
      acc0[0] += w0 * a.x; acc0[1] += w0 * a.y;
      acc1[0] += w1 * b.x; acc1[1] += w1 * b.y;
    }
  }
  if (e < e1) {
    const int   c = edge_col[e];
    const float w = edge_w[e];
    const float* s = feat + (size_t)c * F + lane * VPL;
    if (VPL == 4) {
      v4f a = *(const v4f*)s;
      acc0[0] += w * a.x; acc0[1] += w * a.y;
      acc0[2] += w * a.z; acc0[3] += w * a.w;
    } else {
      v2f a = *(const v2f*)s;
      acc0[0] += w * a.x; acc0[1] += w * a.y;
    }
  }

  float* dst = out + (size_t)node * F + lane * VPL;
  if (VPL == 4) {
    v4f r;
    r.x = acc0[0] + acc1[0]; r.y = acc0[1] + acc1[1];
    r.z = acc0[2] + acc1[2]; r.w = acc0[3] + acc1[3];
    if (RELU) {
      r.x = fmaxf(r.x, 0.f); r.y = fmaxf(r.y, 0.f);
      r.z = fmaxf(r.z, 0.f); r.w = fmaxf(r.w, 0.f);
    }
    *(v4f*)dst = r;
  } else {
    v2f r;
    r.x = acc0[0] + acc1[0]; r.y = acc0[1] + acc1[1];
    if (RELU) { r.x = fmaxf(r.x, 0.f); r.y = fmaxf(r.y, 0.f); }
    *(v2f*)dst = r;
  }
}

// ---------------------------------------------------------------------------
// Launch: x@w0 -> spmm+relu -> h@w1 -> spmm -> out
// ---------------------------------------------------------------------------
extern "C" void kernel_launch(void* const* d_in, const int* in_sizes, int n_in,
                              void* d_out, int out_size, void* d_ws, size_t ws_size,
                              hipStream_t stream) {
  constexpr int IN_F = 128, HID = 128, OUT_F = 64;

  const float* x   = (const float*)d_in[0];
  const int*   er  = (const int*)d_in[1];
  const int*   ec  = (const int*)d_in[2];
  const float* ew  = (const float*)d_in[3];
  const float* w0  = (const float*)d_in[4];
  const float* w1  = (const float*)d_in[5];
  float* out = (float*)d_out;

  const int n_nodes = in_sizes[0] / IN_F;   // 50000
  const int n_edges = in_sizes[1];          // 800000

  // Workspace layout: bufA (nodes*128 f32, reused for hw1), bufB (nodes*128),
  // row_ptr (nodes+1 ints).
  char* ws = (char*)d_ws;
  float* bufA = (float*)ws;
  float* bufB = (float*)(ws + (size_t)n_nodes * HID * sizeof(float));
  int*   rp   = (int*)(ws + 2 * (size_t)n_nodes * HID * sizeof(float));

  // 1) CSR row pointers
  {
    int threads = 256;
    int blocks = (n_nodes + 1 + threads - 1) / threads;
    gcn_rowptr_kernel<<<blocks, threads, 0, stream>>>(er, rp, n_nodes, n_edges);
  }
  // 2) xw0 = x @ w0  [n,128]
  {
    int blocks = (n_nodes + 63) / 64;
    gcn_gemm_wmma_kernel<HID><<<blocks, 256, 0, stream>>>(x, w0, bufA, n_nodes);
  }
  // 3) h = relu(spmm(xw0))  [n,128]
  {
    int blocks = (n_nodes + 7) / 8;
    gcn_spmm_kernel<HID, true><<<blocks, 256, 0, stream>>>(bufA, rp, ec, ew, bufB, n_nodes);
  }
  // 4) hw1 = h @ w1  [n,64]  (reuse bufA)
  {
    int blocks = (n_nodes + 63) / 64;
    gcn_gemm_wmma_kernel<OUT_F><<<blocks, 256, 0, stream>>>(bufB, w1, bufA, n_nodes);
  }
  // 5) logits = spmm(hw1)  [n,64]
  {
    int blocks = (n_nodes + 7) / 8;
    gcn_spmm_kernel<OUT_F, false><<<blocks, 256, 0, stream>>>(bufA, rp, ec, ew, out, n_nodes);
  }
}